// GAT_4681514352902
// MI455X (gfx1250) — compile-verified
//
#include <hip/hip_runtime.h>
#include <math.h>

// Problem constants (match reference)
#define N_NODES 50000
#define N_EDGES 1600000
#define F_IN    128
#define DIM_H   32
#define DIM_OUT 32
#define NEG_SLOPE 0.2f
#define EPS_DEN 1e-16f

typedef __attribute__((ext_vector_type(2))) float v2f;
typedef __attribute__((ext_vector_type(8))) float v8f;

// -------- order-preserving float <-> uint encoding for atomic max --------
__device__ __forceinline__ unsigned ordf(float f) {
    unsigned u = __float_as_uint(f);
    return u ^ ((unsigned)((int)u >> 31) | 0x80000000u);
}
__device__ __forceinline__ float unordf(unsigned o) {
    unsigned u = (o & 0x80000000u) ? (o ^ 0x80000000u) : ~o;
    return __uint_as_float(u);
}
#define ORD_NEG_INF 0x007FFFFFu  // ordf(-inf)

__device__ __forceinline__ float leaky(float v) {
    return v > 0.0f ? v : NEG_SLOPE * v;
}

// -------- init scratch (must run every call: graph replay determinism) ----
__global__ void gat_init(float* __restrict__ accum, float* __restrict__ denom,
                         unsigned* __restrict__ emax) {
    int i = blockIdx.x * blockDim.x + threadIdx.x;
    if (i < N_NODES * DIM_H) accum[i] = 0.0f;
    if (i < N_NODES) { denom[i] = 0.0f; emax[i] = ORD_NEG_INF; }
}

// -------- fold W_dst @ att_dst into a 128-vector --------------------------
__global__ void gat_fold_wd(const float* __restrict__ Wdst,
                            const float* __restrict__ att_dst,
                            float* __restrict__ w_d) {
    int k = threadIdx.x;  // 0..127
    float s = 0.0f;
#pragma unroll
    for (int h = 0; h < DIM_H; ++h) s += Wdst[k * DIM_H + h] * att_dst[h];
    w_d[k] = s;
}

// -------- h_src = x @ W_src via V_WMMA_F32_16X16X4_F32 --------------------
// W_src staged once per block in LDS; one wave computes a 16x32 tile.
__global__ void gat_gemm_hsrc(const float* __restrict__ x,
                              const float* __restrict__ Wsrc,
                              float* __restrict__ h_src) {
    __shared__ float Wl[F_IN * DIM_H];  // 16 KB
    {   // cooperative float4 stage: 1024 float4s over 256 threads
        const float4* ws4 = (const float4*)Wsrc;
        float4* wl4 = (float4*)Wl;
#pragma unroll
        for (int i = threadIdx.x; i < (F_IN * DIM_H) / 4; i += 256)
            wl4[i] = ws4[i];
    }
    __syncthreads();

    const int wave = (blockIdx.x * blockDim.x + threadIdx.x) >> 5;
    const int lane = threadIdx.x & 31;
    const int row0 = wave * 16;
    if (row0 >= N_NODES) return;          // wave-uniform: EXEC full in live waves
    const int half = lane >> 4;           // 0: K offset 0, 1: K offset +2
    const int l = lane & 15;
    const long arow = (long)(row0 + l) * F_IN;
    v8f c0 = {}; v8f c1 = {};
    for (int k = 0; k < F_IN; k += 4) {
        const int ka = k + half * 2;
        v2f a; a.x = x[arow + ka]; a.y = x[arow + ka + 1];
        v2f b0, b1;
        b0.x = Wl[(ka)     * DIM_H + l];
        b0.y = Wl[(ka + 1) * DIM_H + l];
        b1.x = Wl[(ka)     * DIM_H + 16 + l];
        b1.y = Wl[(ka + 1) * DIM_H + 16 + l];
        c0 = __builtin_amdgcn_wmma_f32_16x16x4_f32(false, a, false, b0, (short)0, c0, false, false);
        c1 = __builtin_amdgcn_wmma_f32_16x16x4_f32(false, a, false, b1, (short)0, c1, false, false);
    }
#pragma unroll
    for (int r = 0; r < 8; ++r) {
        const int m = r + half * 8;
        h_src[(long)(row0 + m) * DIM_H + l]      = c0[r];
        h_src[(long)(row0 + m) * DIM_H + 16 + l] = c1[r];
    }
}

// -------- per-node logits: a_src = h_src.att_src ; a_dst = x.w_d ----------
__global__ void gat_node_logits(const float* __restrict__ x,
                                const float* __restrict__ h_src,
                                const float* __restrict__ att_src,
                                const float* __restrict__ w_d,
                                float* __restrict__ a_src,
                                float* __restrict__ a_dst) {
    int i = blockIdx.x * blockDim.x + threadIdx.x;
    if (i >= N_NODES) return;
    const float4* hr = (const float4*)(h_src + (long)i * DIM_H);
    const float4* as = (const float4*)att_src;
    float s = 0.0f;
#pragma unroll
    for (int q = 0; q < DIM_H / 4; ++q) {
        float4 h = hr[q], a = as[q];
        s += h.x * a.x + h.y * a.y + h.z * a.z + h.w * a.w;
    }
    a_src[i] = s;
    const float4* xr = (const float4*)(x + (long)i * F_IN);
    const float4* wd = (const float4*)w_d;
    float d = 0.0f;
#pragma unroll
    for (int q = 0; q < F_IN / 4; ++q) {
        float4 xv = xr[q], wv = wd[q];
        d += xv.x * wv.x + xv.y * wv.y + xv.z * wv.z + xv.w * wv.w;
    }
    a_dst[i] = d;
}

// -------- edge prep: eval[e] = leaky(...); segment max ---------------------
__global__ void gat_edge_prep(const long long* __restrict__ ei,
                              const float* __restrict__ a_src,
                              const float* __restrict__ a_dst,
                              float* __restrict__ eval,
                              unsigned* __restrict__ emax) {
    int e = blockIdx.x * blockDim.x + threadIdx.x;
    if (e >= N_EDGES) return;
    const int s = (int)ei[e];
    const int d = (int)ei[N_EDGES + e];
    const float v = leaky(a_src[s] + a_dst[d]);
    eval[e] = v;
    atomicMax(&emax[d], ordf(v));
}

// -------- segment sum of exp(e - max) -------------------------------------
__global__ void gat_edge_sum(const long long* __restrict__ ei,
                             const float* __restrict__ eval,
                             const unsigned* __restrict__ emax,
                             float* __restrict__ denom) {
    int e = blockIdx.x * blockDim.x + threadIdx.x;
    if (e >= N_EDGES) return;
    const int d = (int)ei[N_EDGES + e];
    const float w = expf(eval[e] - unordf(emax[d]));
    atomicAdd(&denom[d], w);
}

// -------- weighted scatter: accum[d] += alpha * h_src[s] ------------------
// 8 lanes per edge; leader lane computes alpha once, broadcast via shuffle.
__global__ void gat_scatter(const long long* __restrict__ ei,
                            const float* __restrict__ eval,
                            const unsigned* __restrict__ emax,
                            const float* __restrict__ denom,
                            const float* __restrict__ h_src,
                            float* __restrict__ accum) {
    int t = blockIdx.x * blockDim.x + threadIdx.x;
    int e = t >> 3;
    if (e >= N_EDGES) return;
    const int lane = threadIdx.x & 31;
    const int ch = lane & 7;
    float alpha_l = 0.0f;
    int s_l = 0, d_l = 0;
    if (ch == 0) {  // one leader per 8-lane group
        s_l = (int)ei[e];
        d_l = (int)ei[N_EDGES + e];
        alpha_l = expf(eval[e] - unordf(emax[d_l])) / (denom[d_l] + EPS_DEN);
    }
    const int src_lane = lane & ~7;
    const float alpha = __shfl(alpha_l, src_lane, 32);
    const int s = __shfl(s_l, src_lane, 32);
    const int d = __shfl(d_l, src_lane, 32);
    const float4 hv = *(const float4*)(h_src + (long)s * DIM_H + ch * 4);
    float* dp = accum + (long)d * DIM_H + ch * 4;
    atomicAdd(dp + 0, alpha * hv.x);
    atomicAdd(dp + 1, alpha * hv.y);
    atomicAdd(dp + 2, alpha * hv.z);
    atomicAdd(dp + 3, alpha * hv.w);
}

// -------- final: out = relu(accum + bias) @ W_lin + b_lin (WMMA f32) ------
__global__ void gat_final(const float* __restrict__ accum,
                          const float* __restrict__ bias,
                          const float* __restrict__ Wlin,
                          const float* __restrict__ blin,
                          float* __restrict__ out) {
    __shared__ float Wl[DIM_H * DIM_OUT];  // 4 KB
    {
        const float4* ws4 = (const float4*)Wlin;
        float4* wl4 = (float4*)Wl;
        if (threadIdx.x < (DIM_H * DIM_OUT) / 4) wl4[threadIdx.x] = ws4[threadIdx.x];
    }
    __syncthreads();

    const int wave = (blockIdx.x * blockDim.x + threadIdx.x) >> 5;
    const int lane = threadIdx.x & 31;
    const int row0 = wave * 16;
    if (row0 >= N_NODES) return;          // wave-uniform exit
    const int half = lane >> 4;
    const int l = lane & 15;
    const long arow = (long)(row0 + l) * DIM_H;
    v8f c0 = {}; v8f c1 = {};
#pragma unroll
    for (int k = 0; k < DIM_H; k += 4) {
        const int ka = k + half * 2;
        v2f a;
        a.x = fmaxf(accum[arow + ka]     + bias[ka],     0.0f);
        a.y = fmaxf(accum[arow + ka + 1] + bias[ka + 1], 0.0f);
        v2f b0, b1;
        b0.x = Wl[(ka)     * DIM_OUT + l];
        b0.y = Wl[(ka + 1) * DIM_OUT + l];
        b1.x = Wl[(ka)     * DIM_OUT + 16 + l];
        b1.y = Wl[(ka + 1) * DIM_OUT + 16 + l];
        c0 = __builtin_amdgcn_wmma_f32_16x16x4_f32(false, a, false, b0, (short)0, c0, false, false);
        c1 = __builtin_amdgcn_wmma_f32_16x16x4_f32(false, a, false, b1, (short)0, c1, false, false);
    }
    const float bl0 = blin[l];
    const float bl1 = blin[16 + l];
#pragma unroll
    for (int r = 0; r < 8; ++r) {
        const int m = r + half * 8;
        out[(long)(row0 + m) * DIM_OUT + l]      = c0[r] + bl0;
        out[(long)(row0 + m) * DIM_OUT + 16 + l] = c1[r] + bl1;
    }
}

extern "C" void kernel_launch(void* const* d_in, const int* in_sizes, int n_in,
                              void* d_out, int out_size, void* d_ws, size_t ws_size,
                              hipStream_t stream) {
    const float*     x       = (const float*)d_in[0];
    const long long* ei      = (const long long*)d_in[1];  // int64 [2, E]
    const float*     Wsrc    = (const float*)d_in[2];
    const float*     Wdst    = (const float*)d_in[3];
    const float*     att_src = (const float*)d_in[4];
    const float*     att_dst = (const float*)d_in[5];
    const float*     bias    = (const float*)d_in[6];
    const float*     Wlin    = (const float*)d_in[7];
    const float*     blin    = (const float*)d_in[8];
    float* out = (float*)d_out;

    // workspace layout (~20.0 MB)
    float*    h_src = (float*)d_ws;                    // N*32
    float*    accum = h_src + (long)N_NODES * DIM_H;   // N*32
    float*    eval  = accum + (long)N_NODES * DIM_H;   // E
    float*    a_src = eval + N_EDGES;                  // N
    float*    a_dst = a_src + N_NODES;                 // N
    float*    denom = a_dst + N_NODES;                 // N
    unsigned* emax  = (unsigned*)(denom + N_NODES);    // N
    float*    w_d   = (float*)(emax + N_NODES);        // 128

    const int T = 256;
    const int waves_gemm = (N_NODES + 15) / 16;                 // 3125
    const int blocks_gemm = (waves_gemm * 32 + T - 1) / T;      // 391

    gat_init<<<(N_NODES * DIM_H + T - 1) / T, T, 0, stream>>>(accum, denom, emax);
    gat_fold_wd<<<1, F_IN, 0, stream>>>(Wdst, att_dst, w_d);
    gat_gemm_hsrc<<<blocks_gemm, T, 0, stream>>>(x, Wsrc, h_src);
    gat_node_logits<<<(N_NODES + T - 1) / T, T, 0, stream>>>(x, h_src, att_src, w_d, a_src, a_dst);
    gat_edge_prep<<<(N_EDGES + T - 1) / T, T, 0, stream>>>(ei, a_src, a_dst, eval, emax);
    gat_edge_sum<<<(N_EDGES + T - 1) / T, T, 0, stream>>>(ei, eval, emax, denom);
    gat_scatter<<<((long)N_EDGES * 8 + T - 1) / T, T, 0, stream>>>(ei, eval, emax, denom, h_src, accum);
    gat_final<<<blocks_gemm, T, 0, stream>>>(accum, bias, Wlin, blin, out);
}